// RoIPool_23622320128427
// MI455X (gfx1250) — compile-verified
//
#include <hip/hip_runtime.h>

#define PPH 7
#define PPW 7
#define NROIS 256
#define NC 256
#define FH 38
#define FW 38
#define PLANE (FH * FW)          // 1444 floats per channel plane
#define CH_PER_BLK 8             // channels staged per block
#define THREADS 256              // 8 wave32s
#define OUTS_PER_BLK (CH_PER_BLK * PPH * PPW)   // 392

typedef float v4f __attribute__((ext_vector_type(4)));
typedef int   v4i __attribute__((ext_vector_type(4)));

#if defined(__AMDGCN__) && defined(__gfx1250__) && \
    __has_builtin(__builtin_amdgcn_global_load_async_to_lds_b128)
#define USE_ASYNC_LDS 1
typedef __attribute__((address_space(1))) v4i* async_gptr_t;  // global src
typedef __attribute__((address_space(3))) v4i* async_lptr_t;  // LDS dst
#else
#define USE_ASYNC_LDS 0
#endif

struct BinRect { int hs, he, ws, we, cl, ph, pw; };

__device__ __forceinline__ BinRect make_rect(int idx, int x1, int y1,
                                             float bw, float bh) {
  BinRect b;
  b.cl = idx / (PPH * PPW);
  const int bin = idx % (PPH * PPW);
  b.ph = bin / PPW;
  b.pw = bin % PPW;
  b.hs = min(max((int)floorf((float)b.ph * bh) + y1, 0), FH);
  b.he = min(max((int)ceilf((float)(b.ph + 1) * bh) + y1, 0), FH);
  b.ws = min(max((int)floorf((float)b.pw * bw) + x1, 0), FW);
  b.we = min(max((int)ceilf((float)(b.pw + 1) * bw) + x1, 0), FW);
  return b;
}

__device__ __forceinline__ float pool_rect(const float* smem, const BinRect& b) {
  if (b.he <= b.hs || b.we <= b.ws) return 0.0f;   // empty bin -> 0
  float m = -__builtin_huge_valf();
  const float* p = smem + b.cl * PLANE;
  for (int h = b.hs; h < b.he; ++h) {
    const float* row = p + h * FW;
    for (int w = b.ws; w < b.we; ++w) m = fmaxf(m, row[w]);
  }
  return m;
}

__global__ __launch_bounds__(THREADS)
void roipool_gfx1250_kernel(const float* __restrict__ feat,
                            const float* __restrict__ rois,
                            const float* __restrict__ scale_p,
                            float* __restrict__ out) {
  __shared__ float smem[CH_PER_BLK * PLANE];   // 45.125 KB

  const int blocks_per_roi = NC / CH_PER_BLK;  // 32
  const int roi   = blockIdx.x / blocks_per_roi;
  const int chunk = blockIdx.x % blocks_per_roi;
  const int c0    = chunk * CH_PER_BLK;
  const int tid   = threadIdx.x;

  // ---- (1) Kick off DMA: stage 8 contiguous channel planes (45 KB span) ----
  const float* src = feat + (size_t)c0 * PLANE;
  const int nvec = (CH_PER_BLK * PLANE) / 4;   // 2888 x b128 transfers

#if USE_ASYNC_LDS
  for (int i = tid; i < nvec; i += THREADS) {
    __builtin_amdgcn_global_load_async_to_lds_b128(
        (async_gptr_t)(src + 4 * i),
        (async_lptr_t)(smem + 4 * i),
        /*offset=*/0, /*cpol=*/0);
  }
#else
  for (int i = tid; i < nvec; i += THREADS) {
    ((v4f*)smem)[i] = ((const v4f*)src)[i];
  }
#endif

  // ---- (2) Overlap: ROI coords + per-thread bin rectangles while DMA flies ----
  const float scale = scale_p[0];
  const float* r = rois + roi * 5;
  // rintf = round-half-even, matching jnp.round
  const int x1 = (int)rintf(r[1] * scale);
  const int y1 = (int)rintf(r[2] * scale);
  const int x2 = (int)rintf(r[3] * scale);
  const int y2 = (int)rintf(r[4] * scale);
  const float bw = (float)max(x2 - x1 + 1, 1) / (float)PPW;
  const float bh = (float)max(y2 - y1 + 1, 1) / (float)PPH;

  // 392 outputs per block: thread t owns idx=t and (if t<136) idx=t+256
  const BinRect b0 = make_rect(tid, x1, y1, bw, bh);
  const bool has2 = (tid + THREADS) < OUTS_PER_BLK;
  const BinRect b1 = make_rect(has2 ? tid + THREADS : 0, x1, y1, bw, bh);

  // ---- (3) Complete DMA, then pool straight out of LDS ----
#if USE_ASYNC_LDS
#if __has_builtin(__builtin_amdgcn_s_wait_asynccnt)
  __builtin_amdgcn_s_wait_asynccnt(0);
#else
  asm volatile("s_wait_asynccnt 0" ::: "memory");
#endif
#endif
  __syncthreads();

  const size_t obase = (size_t)roi * NC * PPH * PPW + (size_t)c0 * PPH * PPW;
  {
    const float m = pool_rect(smem, b0);
    out[obase + tid] = m;   // idx -> (cl*49 + ph*7 + pw), contiguous per block
  }
  if (has2) {
    const float m = pool_rect(smem, b1);
    out[obase + tid + THREADS] = m;
  }
}

extern "C" void kernel_launch(void* const* d_in, const int* in_sizes, int n_in,
                              void* d_out, int out_size, void* d_ws, size_t ws_size,
                              hipStream_t stream) {
  (void)in_sizes; (void)n_in; (void)out_size; (void)d_ws; (void)ws_size;
  const float* feat  = (const float*)d_in[0];   // (1,256,38,38) f32
  const float* rois  = (const float*)d_in[1];   // (256,5) f32
  const float* scale = (const float*)d_in[2];   // scalar f32
  float* out = (float*)d_out;                   // (256,256,7,7) f32

  const int nblocks = NROIS * (NC / CH_PER_BLK);  // 8192
  roipool_gfx1250_kernel<<<dim3(nblocks), dim3(THREADS), 0, stream>>>(
      feat, rois, scale, out);
}